// ScalableGraphSAGELayer_88373247082992
// MI455X (gfx1250) — compile-verified
//
#include <hip/hip_runtime.h>
#include <math.h>

// Problem constants (from reference)
#define NN      50000
#define EE      800000
#define EPRIME  850000          // E + N self loops (exact multiple of 16)
#define IN_DIM  128
#define OUT_DIM 128
#define ED_DIM  64
#define HEADS   4
#define CHAN    32
#define NEG_SLOPE 0.2f
#define EPS_GN  1e-5f

typedef __attribute__((ext_vector_type(2))) float v2f;
typedef __attribute__((ext_vector_type(8))) float v8f;

// ---- order-preserving float<->uint encoding for atomic segment-max ----
__device__ __forceinline__ unsigned encf(float f) {
  unsigned u = __float_as_uint(f);
  return (u & 0x80000000u) ? ~u : (u | 0x80000000u);
}
__device__ __forceinline__ float decf(unsigned u) {
  return (u & 0x80000000u) ? __uint_as_float(u & 0x7FFFFFFFu)
                           : __uint_as_float(~u);
}

// ---------------------------------------------------------------------
// K1: degree + per-dst edge_attr sums (for self-loop fill_value='mean')
// grid over E*64 elements
__global__ void k_deg_loopsum(const float* __restrict__ edge_attr,
                              const long long* __restrict__ ei,
                              float* __restrict__ loop_attr,
                              float* __restrict__ deg) {
  long long gid = (long long)blockIdx.x * 256 + threadIdx.x;   // < E*64 exact
  int e = (int)(gid >> 6);
  int c = (int)(gid & 63);
  int dst = (int)ei[EE + e];
  atomicAdd(&loop_attr[(size_t)dst * ED_DIM + c], edge_attr[gid]);
  if (c == 0) atomicAdd(&deg[dst], 1.0f);
}

// K2: loop_attr /= max(deg,1)
__global__ void k_loopnorm(float* __restrict__ loop_attr,
                           const float* __restrict__ deg) {
  int gid = blockIdx.x * 256 + threadIdx.x;                    // < N*64 exact
  int n = gid >> 6;
  loop_attr[gid] = loop_attr[gid] / fmaxf(deg[n], 1.0f);
}

// ---------------------------------------------------------------------
// K3: node GEMMs  xl = x@Wl+bl,  xr = x@Wr+br  via V_WMMA_F32_16X16X4_F32
// block = 256 threads = 8 waves; block handles 16 rows x 128 cols;
// wave w owns columns [16w, 16w+16). K-loop of 32 WMMA steps (K=4 each).
__global__ void k_node_gemm(const float* __restrict__ x,
                            const float* __restrict__ Wl,
                            const float* __restrict__ bl,
                            const float* __restrict__ Wr,
                            const float* __restrict__ br,
                            float* __restrict__ xl,
                            float* __restrict__ xr) {
  const int t = threadIdx.x;
  const int wave = t >> 5, lane = t & 31;
  const int half = lane >> 4, m = lane & 15;
  const int r0 = blockIdx.x * 16;                // N = 50000 = 3125*16 exact
  const int col = wave * 16 + m;

  v8f cl = {};
  v8f cr = {};
  #pragma unroll
  for (int k = 0; k < IN_DIM; k += 4) {
    // A frag (16x4): lane<16 -> K=k,k+1 ; lane>=16 -> K=k+2,k+3 ; M=m
    const float* xp = x + (size_t)(r0 + m) * IN_DIM + k + 2 * half;
    v2f a; a.x = xp[0]; a.y = xp[1];
    // B frag (4x16), transpose-symmetric layout: VGPR0 K=k+2*half, VGPR1 K+1
    int kr = k + 2 * half;
    v2f b1; b1.x = Wl[kr * 128 + col]; b1.y = Wl[(kr + 1) * 128 + col];
    v2f b2; b2.x = Wr[kr * 128 + col]; b2.y = Wr[(kr + 1) * 128 + col];
    cl = __builtin_amdgcn_wmma_f32_16x16x4_f32(false, a, false, b1,
                                               (short)0, cl, false, false);
    cr = __builtin_amdgcn_wmma_f32_16x16x4_f32(false, a, false, b2,
                                               (short)0, cr, false, false);
  }
  float bll = bl[col], brr = br[col];
  #pragma unroll
  for (int v = 0; v < 8; ++v) {
    int row = r0 + v + 8 * half;                 // C/D: VGPR v -> M=v / v+8
    xl[(size_t)row * 128 + col] = cl[v] + bll;
    xr[(size_t)row * 128 + col] = cr[v] + brr;
  }
}

// ---------------------------------------------------------------------
// K4: fused edge GEMM ee = eattr@We + GATv2 score + segment-max atomics.
// block = 256 threads = 8 waves; block handles one 16-edge tile.
__global__ void k_edge_score(const float* __restrict__ edge_attr,
                             const long long* __restrict__ ei,
                             const float* __restrict__ loop_attr,
                             const float* __restrict__ xl,
                             const float* __restrict__ xr,
                             const float* __restrict__ We,
                             const float* __restrict__ att,
                             float* __restrict__ score,
                             unsigned* __restrict__ smax) {
  __shared__ float sAttr[16][ED_DIM];   // 4 KB tile of (augmented) edge attrs
  __shared__ int   sSrc[16];
  __shared__ int   sDst[16];
  __shared__ float sPart[8][16];        // per-wave 16-col partial scores

  const int t = threadIdx.x;
  const int base = blockIdx.x * 16;     // E' = 850000 = 53125*16 exact

  { // cooperative fill: thread -> 4 consecutive floats
    int eloc = t >> 4;
    int c4 = (t & 15) << 2;
    int e = base + eloc;
    const float* p = (e < EE) ? (edge_attr + (size_t)e * ED_DIM + c4)
                              : (loop_attr + (size_t)(e - EE) * ED_DIM + c4);
    sAttr[eloc][c4 + 0] = p[0];
    sAttr[eloc][c4 + 1] = p[1];
    sAttr[eloc][c4 + 2] = p[2];
    sAttr[eloc][c4 + 3] = p[3];
  }
  if (t < 16) {
    int e = base + t;
    if (e < EE) { sSrc[t] = (int)ei[e]; sDst[t] = (int)ei[EE + e]; }
    else        { sSrc[t] = sDst[t] = e - EE; }
  }
  __syncthreads();

  const int wave = t >> 5, lane = t & 31;
  const int half = lane >> 4, m = lane & 15;
  const int col = wave * 16 + m;        // output channel (h*32+c == col)

  v8f acc = {};
  #pragma unroll
  for (int kk = 0; kk < 16; ++kk) {     // K = 64 in steps of 4
    int k = kk * 4;
    v2f a; a.x = sAttr[m][k + 2 * half];
           a.y = sAttr[m][k + 2 * half + 1];
    int kr = k + 2 * half;
    v2f b; b.x = We[kr * 128 + col];
           b.y = We[(kr + 1) * 128 + col];
    acc = __builtin_amdgcn_wmma_f32_16x16x4_f32(false, a, false, b,
                                                (short)0, acc, false, false);
  }

  const float attc = att[col];
  #pragma unroll
  for (int v = 0; v < 8; ++v) {
    int eloc = v + 8 * half;            // D: VGPR v -> edge v (lanes<16) / v+8
    float msg = acc[v]
              + xl[(size_t)sSrc[eloc] * 128 + col]
              + xr[(size_t)sDst[eloc] * 128 + col];
    float lr = msg > 0.f ? msg : NEG_SLOPE * msg;
    float tv = lr * attc;
    // reduce over the 16 columns held by this half-wave
    tv += __shfl_xor(tv, 1, 32);
    tv += __shfl_xor(tv, 2, 32);
    tv += __shfl_xor(tv, 4, 32);
    tv += __shfl_xor(tv, 8, 32);
    if (m == 0) sPart[wave][eloc] = tv;
  }
  __syncthreads();

  if (t < 64) {                         // 16 edges x 4 heads
    int eloc = t & 15, h = t >> 4;
    float s = sPart[2 * h][eloc] + sPart[2 * h + 1][eloc];
    int e = base + eloc;
    score[(size_t)e * HEADS + h] = s;
    atomicMax(&smax[(size_t)sDst[eloc] * HEADS + h], encf(s));
  }
}

// ---------------------------------------------------------------------
// K5: ex = exp(score - smax[dst]); denom[dst] += ex  (ex stored in place)
__global__ void k_exp_denom(const long long* __restrict__ ei,
                            const unsigned* __restrict__ smax,
                            float* __restrict__ score,
                            float* __restrict__ denom) {
  int gid = blockIdx.x * 256 + threadIdx.x;
  if (gid >= EPRIME * HEADS) return;
  int e = gid >> 2, h = gid & 3;
  int dst = (e < EE) ? (int)ei[EE + e] : (e - EE);
  float ex = expf(score[gid] - decf(smax[(size_t)dst * HEADS + h]));
  score[gid] = ex;
  atomicAdd(&denom[(size_t)dst * HEADS + h], ex);
}

// K6: out_accum[dst, c] += ex[e, h(c)] * xl[src, c]
__global__ void k_aggregate(const float* __restrict__ ex,
                            const long long* __restrict__ ei,
                            const float* __restrict__ xl,
                            float* __restrict__ out_accum) {
  long long gid = (long long)blockIdx.x * 256 + threadIdx.x;   // < E'*128 exact
  int e = (int)(gid >> 7);
  int c = (int)(gid & 127);
  int src, dst;
  if (e < EE) { src = (int)ei[e]; dst = (int)ei[EE + e]; }
  else        { src = dst = e - EE; }
  float a = ex[(size_t)e * HEADS + (c >> 5)];
  atomicAdd(&out_accum[(size_t)dst * 128 + c], a * xl[(size_t)src * 128 + c]);
}

// ---------------------------------------------------------------------
// K7: per-channel sum / sum-of-squares of out = accum/denom + bias
__global__ void k_colstats(const float* __restrict__ out_accum,
                           const float* __restrict__ denom,
                           const float* __restrict__ bias_out,
                           float* __restrict__ sums /*[256]: sum|sumsq*/) {
  __shared__ float sh[512];
  int t = threadIdx.x;
  int c = t & 127;
  int r = t >> 7;                       // 0 or 1
  float s = 0.f, s2 = 0.f;
  for (int n = blockIdx.x * 2 + r; n < NN; n += gridDim.x * 2) {
    float val = out_accum[(size_t)n * 128 + c]
                  / (denom[(size_t)n * HEADS + (c >> 5)] + 1e-16f)
              + bias_out[c];
    s += val; s2 += val * val;
  }
  sh[t] = s; sh[256 + t] = s2;
  __syncthreads();
  if (t < 128) {
    atomicAdd(&sums[c],       sh[t]       + sh[t + 128]);
    atomicAdd(&sums[128 + c], sh[256 + t] + sh[256 + t + 128]);
  }
}

// K8: GraphNorm + ReLU
__global__ void k_finalize(const float* __restrict__ out_accum,
                           const float* __restrict__ denom,
                           const float* __restrict__ bias_out,
                           const float* __restrict__ sums,
                           const float* __restrict__ gnw,
                           const float* __restrict__ gnb,
                           const float* __restrict__ gms,
                           float* __restrict__ out) {
  int gid = blockIdx.x * 256 + threadIdx.x;                    // < N*128 exact
  int c = gid & 127;
  int n = gid >> 7;
  float val = out_accum[gid] / (denom[(size_t)n * HEADS + (c >> 5)] + 1e-16f)
            + bias_out[c];
  const float invN = 1.0f / (float)NN;
  float mean = sums[c] * invN;
  float m2   = sums[128 + c] * invN;
  float ms   = mean * gms[c];
  float var  = m2 - 2.f * ms * mean + ms * ms;   // E[(out-mean*gms)^2]
  float y = gnw[c] * (val - ms) * rsqrtf(var + EPS_GN) + gnb[c];
  out[gid] = fmaxf(y, 0.f);
}

// ---------------------------------------------------------------------
extern "C" void kernel_launch(void* const* d_in, const int* in_sizes, int n_in,
                              void* d_out, int out_size, void* d_ws, size_t ws_size,
                              hipStream_t stream) {
  const float*     x         = (const float*)d_in[0];
  const long long* ei        = (const long long*)d_in[1];   // int64 [2,E]
  const float*     edge_attr = (const float*)d_in[2];
  const float*     Wl        = (const float*)d_in[3];
  const float*     bl        = (const float*)d_in[4];
  const float*     Wr        = (const float*)d_in[5];
  const float*     br        = (const float*)d_in[6];
  const float*     We        = (const float*)d_in[7];
  const float*     att       = (const float*)d_in[8];
  const float*     bias_out  = (const float*)d_in[9];
  const float*     gnw       = (const float*)d_in[10];
  const float*     gnb       = (const float*)d_in[11];
  const float*     gms       = (const float*)d_in[12];
  float* out = (float*)d_out;

  // workspace layout (zero-prefix first)
  size_t off = 0;
  auto alloc = [&](size_t bytes) {
    size_t o = off; off = (off + bytes + 255) & ~(size_t)255; return o;
  };
  char* ws = (char*)d_ws;
  size_t o_deg   = alloc((size_t)NN * 4);
  size_t o_smax  = alloc((size_t)NN * HEADS * 4);
  size_t o_denom = alloc((size_t)NN * HEADS * 4);
  size_t o_sums  = alloc(256 * 4);
  size_t o_loop  = alloc((size_t)NN * ED_DIM * 4);
  size_t o_oacc  = alloc((size_t)NN * 128 * 4);
  size_t zeroBytes = off;
  size_t o_xl    = alloc((size_t)NN * 128 * 4);
  size_t o_xr    = alloc((size_t)NN * 128 * 4);
  size_t o_score = alloc((size_t)EPRIME * HEADS * 4);
  (void)ws_size; (void)in_sizes; (void)n_in; (void)out_size;

  float*    deg   = (float*)(ws + o_deg);
  unsigned* smax  = (unsigned*)(ws + o_smax);
  float*    denom = (float*)(ws + o_denom);
  float*    sums  = (float*)(ws + o_sums);
  float*    loopA = (float*)(ws + o_loop);
  float*    oacc  = (float*)(ws + o_oacc);
  float*    xl    = (float*)(ws + o_xl);
  float*    xr    = (float*)(ws + o_xr);
  float*    score = (float*)(ws + o_score);

  hipMemsetAsync(d_ws, 0, zeroBytes, stream);   // smax enc: 0 < enc(any float)

  k_deg_loopsum<<<(EE * 64) / 256, 256, 0, stream>>>(edge_attr, ei, loopA, deg);
  k_loopnorm<<<(NN * 64) / 256, 256, 0, stream>>>(loopA, deg);
  k_node_gemm<<<NN / 16, 256, 0, stream>>>(x, Wl, bl, Wr, br, xl, xr);
  k_edge_score<<<EPRIME / 16, 256, 0, stream>>>(edge_attr, ei, loopA, xl, xr,
                                                We, att, score, smax);
  k_exp_denom<<<(EPRIME * HEADS + 255) / 256, 256, 0, stream>>>(ei, smax,
                                                                score, denom);
  k_aggregate<<<(int)(((long long)EPRIME * 128) / 256), 256, 0, stream>>>(
      score, ei, xl, oacc);
  k_colstats<<<256, 256, 0, stream>>>(oacc, denom, bias_out, sums);
  k_finalize<<<(NN * 128) / 256, 256, 0, stream>>>(oacc, denom, bias_out, sums,
                                                   gnw, gnb, gms, out);
}